// FGL_82480551952955
// MI455X (gfx1250) — compile-verified
//
#include <hip/hip_runtime.h>

typedef __attribute__((ext_vector_type(2))) float v2f;
typedef __attribute__((ext_vector_type(8))) float v8f;

#define BATCH      2
#define N_IN       50000
#define N_OUT      12500
#define NNZ_       500000
#define IN_C       256
#define OUT_C      256

// ===========================================================================
// WMMA fp32 GEMM:  out[r, :] = A[r, :] @ W  (+ bias[r % N_OUT, :] if HAS_BIAS)
// Each wave32 computes ONE 16-row tile x FOUR 16-col tiles (64 output cols):
// one A fragment feeds 4 independent V_WMMA_F32_16X16X4_F32 per K-step,
// cutting A traffic 4x. 8 waves / 256-thread block.
// fp32 WMMA VGPR layouts (ISA 7.12.2):
//   A 16x4 : lanes 0-15 -> M=lane, {K=k0,k0+1}; lanes 16-31 -> {K=k0+2,k0+3}
//   B 4x16 : lanes 0-15 -> N=lane, {K=k0,k0+1}; lanes 16-31 -> {K=k0+2,k0+3}
//   C 16x16: VGPR i -> row i + 8*half, col = lane & 15
// ===========================================================================
template <bool HAS_BIAS>
__global__ void FGL_gemm_wmma(const float* __restrict__ A,
                              const float* __restrict__ w,
                              const float* __restrict__ bias,
                              float*       __restrict__ out,
                              int rowsTotal) {
    const int lane = threadIdx.x & 31;
    const int wave = threadIdx.x >> 5;

    const int numRowTiles = (rowsTotal + 15) / 16;
    const int tile = blockIdx.x * 8 + wave;
    if (tile >= numRowTiles * 4) return;               // wave-uniform exit

    const int tileRow = tile >> 2;
    const int group   = tile & 3;                      // which 64-col slab

    const int half = lane >> 4;                        // K pair select
    const int lidx = lane & 15;

    // A operand (row clamped so EXEC stays all-1s at the WMMAs)
    const int mRow = tileRow * 16 + lidx;
    const int aRow = (mRow < rowsTotal) ? mRow : (rowsTotal - 1);
    const float* aPtr = A + (size_t)aRow * IN_C + 2 * half;

    // B operands: W[k][n], 4 col tiles at n0, n0+16, n0+32, n0+48
    const int nCol0 = group * 64 + lidx;
    const float* bPtr = w + (size_t)(2 * half) * OUT_C + nCol0;

    v8f acc0 = {}, acc1 = {}, acc2 = {}, acc3 = {};
    #pragma unroll 4
    for (int k = 0; k < IN_C; k += 4) {
        v2f a = *(const v2f*)(aPtr + k);               // shared A fragment
        const float* bk = bPtr + (size_t)k * OUT_C;
        v2f b0, b1, b2, b3;
        b0.x = bk[0];   b0.y = bk[OUT_C];
        b1.x = bk[16];  b1.y = bk[OUT_C + 16];
        b2.x = bk[32];  b2.y = bk[OUT_C + 32];
        b3.x = bk[48];  b3.y = bk[OUT_C + 48];
        acc0 = __builtin_amdgcn_wmma_f32_16x16x4_f32(false, a, false, b0, (short)0, acc0, false, false);
        acc1 = __builtin_amdgcn_wmma_f32_16x16x4_f32(false, a, false, b1, (short)0, acc1, false, false);
        acc2 = __builtin_amdgcn_wmma_f32_16x16x4_f32(false, a, false, b2, (short)0, acc2, false, false);
        acc3 = __builtin_amdgcn_wmma_f32_16x16x4_f32(false, a, false, b3, (short)0, acc3, false, false);
    }

    // Epilogue. VGPR i -> row rbase + i + 8*half.
    const int  rbase = tileRow * 16;
    const bool full  = (rbase + 16 <= rowsTotal);

    auto storeTile = [&](const v8f& acc, int o) {
        if (full) {
            #pragma unroll
            for (int i = 0; i < 8; ++i) {
                const int r = rbase + i + 8 * half;
                float v = acc[i];
                if (HAS_BIAS) {
                    const int n = (r >= N_OUT) ? r - N_OUT : r;   // r % N_OUT
                    v += bias[(size_t)n * OUT_C + o];
                }
                out[(size_t)r * OUT_C + o] = v;
            }
        } else {
            #pragma unroll
            for (int i = 0; i < 8; ++i) {
                const int r = rbase + i + 8 * half;
                if (r < rowsTotal) {
                    float v = acc[i];
                    if (HAS_BIAS) {
                        const int n = (r >= N_OUT) ? r - N_OUT : r;
                        v += bias[(size_t)n * OUT_C + o];
                    }
                    out[(size_t)r * OUT_C + o] = v;
                }
            }
        }
    };
    storeTile(acc0, nCol0);
    storeTile(acc1, nCol0 + 16);
    storeTile(acc2, nCol0 + 32);
    storeTile(acc3, nCol0 + 48);
}

// ===========================================================================
// CSR build: histogram -> single-block scan -> scatter permutation
// ===========================================================================
__global__ void FGL_hist(const int* __restrict__ rows, int* __restrict__ cnt) {
    const int e = blockIdx.x * blockDim.x + threadIdx.x;
    if (e < NNZ_) atomicAdd(&cnt[rows[e]], 1);
}

__global__ void FGL_scan(const int* __restrict__ cnt_in,
                         int* __restrict__ rowStart,
                         int* __restrict__ cursor) {
    __shared__ int sdata[1024];
    __shared__ int s_running;
    const int tid = threadIdx.x;
    if (tid == 0) s_running = 0;
    __syncthreads();

    for (int base = 0; base < N_OUT; base += 1024) {
        const int i = base + tid;
        const int v = (i < N_OUT) ? cnt_in[i] : 0;
        sdata[tid] = v;
        __syncthreads();
        for (int off = 1; off < 1024; off <<= 1) {      // Hillis-Steele
            const int t = (tid >= off) ? sdata[tid - off] : 0;
            __syncthreads();
            sdata[tid] += t;
            __syncthreads();
        }
        const int excl = sdata[tid] - v;
        if (i < N_OUT) {
            rowStart[i] = s_running + excl;
            cursor[i]   = s_running + excl;
        }
        const int total = sdata[1023];
        __syncthreads();
        if (tid == 0) s_running += total;
        __syncthreads();
    }
    if (tid == 0) rowStart[N_OUT] = NNZ_;
}

__global__ void FGL_scatter(const int*   __restrict__ rows,
                            const int*   __restrict__ cols,
                            const float* __restrict__ vals,
                            int*   __restrict__ cursor,
                            int*   __restrict__ permCols,
                            float* __restrict__ permVals) {
    const int e = blockIdx.x * blockDim.x + threadIdx.x;
    if (e >= NNZ_) return;
    const int p = atomicAdd(&cursor[rows[e]], 1);
    permCols[p] = cols[e];
    permVals[p] = vals[e];
}

// ===========================================================================
// CSR gather-SpMM with fused bias:  out[b,r,:] = sum_j v_j * xw[b,c_j,:] + bias
// One block per (row, batch); thread = channel. Segment (col,val) entries are
// staged into LDS with one coalesced burst, then broadcast via ds_load.
// All traffic L2-resident (xw 102 MB + out 25.6 MB < 192 MB L2), no fp atomics.
// ===========================================================================
__global__ void FGL_csr_spmm(const float* __restrict__ xw,
                             const int*   __restrict__ rowStart,
                             const int*   __restrict__ permCols,
                             const float* __restrict__ permVals,
                             const float* __restrict__ bias,
                             float*       __restrict__ out) {
    __shared__ int   sCols[256];
    __shared__ float sVals[256];

    const int r = blockIdx.x;
    const int b = blockIdx.y;
    const int c = threadIdx.x;

    const int jb = rowStart[r];
    const int je = rowStart[r + 1];
    const float* xwb = xw + (size_t)b * N_IN * OUT_C;

    float acc = 0.f;
    for (int base = jb; base < je; base += 256) {
        const int len = min(je - base, 256);
        if (c < len) {
            sCols[c] = permCols[base + c];
            sVals[c] = permVals[base + c];
        }
        __syncthreads();
        for (int j = 0; j < len; ++j) {
            acc = fmaf(sVals[j], xwb[(size_t)sCols[j] * OUT_C + c], acc);
        }
        __syncthreads();
    }
    out[((size_t)b * N_OUT + r) * OUT_C + c] = acc + bias[(size_t)r * OUT_C + c];
}

// ===========================================================================
// Fallback path (small workspace): scatter-atomic SpMM into ax, then GEMM.
// ===========================================================================
__global__ void FGL_spmm_scatter(const float* __restrict__ x,
                                 const int*   __restrict__ rows,
                                 const int*   __restrict__ cols,
                                 const float* __restrict__ vals,
                                 float*       __restrict__ ax) {
    const int e = blockIdx.x;
    const int b = blockIdx.y;
    const int c = threadIdx.x;
    const float xv = x[((size_t)b * N_IN + cols[e]) * IN_C + c];
    unsafeAtomicAdd(&ax[((size_t)b * N_OUT + rows[e]) * IN_C + c], xv * vals[e]);
}

// ===========================================================================
static inline size_t alignUp(size_t v, size_t a) { return (v + a - 1) & ~(a - 1); }

extern "C" void kernel_launch(void* const* d_in, const int* in_sizes, int n_in,
                              void* d_out, int out_size, void* d_ws, size_t ws_size,
                              hipStream_t stream) {
    const float* x    = (const float*)d_in[0];   // [B, N_IN, IN_C]
    const int*   rows = (const int*)  d_in[1];   // [NNZ]
    const int*   cols = (const int*)  d_in[2];   // [NNZ]
    const float* vals = (const float*)d_in[3];   // [NNZ]
    const float* w    = (const float*)d_in[4];   // [IN_C, OUT_C]
    const float* bias = (const float*)d_in[5];   // [N_OUT, OUT_C]
    float*       out  = (float*)d_out;           // [B, N_OUT, OUT_C]
    char*        ws   = (char*)d_ws;

    // ---- Plan A workspace: xw | rowStart | cursor | permCols | permVals ----
    const size_t xwBytes = (size_t)BATCH * N_IN * OUT_C * sizeof(float); // 102.4 MB
    const size_t off_rs  = alignUp(xwBytes, 256);
    const size_t off_cur = alignUp(off_rs + (N_OUT + 1) * sizeof(int), 256);
    const size_t off_pc  = alignUp(off_cur + N_OUT * sizeof(int), 256);
    const size_t off_pv  = alignUp(off_pc + (size_t)NNZ_ * sizeof(int), 256);
    const size_t needA   = off_pv + (size_t)NNZ_ * sizeof(float);

    if (ws_size >= needA) {
        // ---- Plan A: out = CSR-gather( x @ W ) + bias ; zero fp atomics ----
        float* xw       = (float*)ws;
        int*   rowStart = (int*)  (ws + off_rs);
        int*   cursor   = (int*)  (ws + off_cur);
        int*   permCols = (int*)  (ws + off_pc);
        float* permVals = (float*)(ws + off_pv);

        hipMemsetAsync(cursor, 0, N_OUT * sizeof(int), stream);
        FGL_hist<<<(NNZ_ + 255) / 256, 256, 0, stream>>>(rows, cursor);
        FGL_scan<<<1, 1024, 0, stream>>>(cursor, rowStart, cursor);
        FGL_scatter<<<(NNZ_ + 255) / 256, 256, 0, stream>>>(
            rows, cols, vals, cursor, permCols, permVals);

        // xw = x @ W   (100000 rows = 6250 row tiles, exactly divisible)
        const int rowsXW  = BATCH * N_IN;
        const int tilesXW = ((rowsXW + 15) / 16) * 4;       // 4 col-slabs
        FGL_gemm_wmma<false><<<(tilesXW + 7) / 8, 256, 0, stream>>>(
            x, w, nullptr, xw, rowsXW);

        dim3 gS(N_OUT, BATCH);
        FGL_csr_spmm<<<gS, OUT_C, 0, stream>>>(
            xw, rowStart, permCols, permVals, bias, out);
    } else {
        // ---- Plan B fallback: atomic scatter into ax, then GEMM + bias ----
        float* ax = (float*)ws;                              // [B*N_OUT, IN_C]
        const size_t axBytes = (size_t)BATCH * N_OUT * IN_C * sizeof(float);
        hipMemsetAsync(ax, 0, axBytes, stream);

        dim3 g1(NNZ_, BATCH);
        FGL_spmm_scatter<<<g1, IN_C, 0, stream>>>(x, rows, cols, vals, ax);

        const int rowsAX  = BATCH * N_OUT;                   // 25000
        const int tilesAX = ((rowsAX + 15) / 16) * 4;
        FGL_gemm_wmma<true><<<(tilesAX + 7) / 8, 256, 0, stream>>>(
            ax, w, bias, out, rowsAX);
    }
}